// BilinearSeqAttnAction2_11175504904502
// MI455X (gfx1250) — compile-verified
//
#include <hip/hip_runtime.h>
#include <hip/hip_bf16.h>
#include <math.h>

// CDNA5 (gfx1250) wave32. One f32 WMMA: V_WMMA_F32_16X16X4_F32.
typedef __attribute__((ext_vector_type(2))) float v2f;
typedef __attribute__((ext_vector_type(8))) float v8f;

#define TPB 256  // 8 waves of 32

// ---------------------------------------------------------------------------
// Phase A: Wy[b,x] = sum_y y[b,y] * weight[actions[b], y, x] + bias[actions[b], x]
// One block per batch. Weight rows read with coalesced float4 (wave covers 512B
// contiguous); total unique weight traffic (<=128MB) is L2-resident on MI455X.
// ---------------------------------------------------------------------------
__global__ void __launch_bounds__(TPB) wy_kernel(
    const float* __restrict__ y, const int* __restrict__ actions,
    const float* __restrict__ weight, const float* __restrict__ bias,
    float* __restrict__ Wy, int L, int X, int Y)
{
    const int b   = blockIdx.x;
    const int tid = threadIdx.x;

    extern __shared__ float ys[];  // Y floats
    for (int i = tid; i < Y; i += TPB) ys[i] = y[(size_t)b * Y + i];
    __syncthreads();

    const int a = actions[b];
    const float* __restrict__ W = weight + (size_t)a * Y * X;

    for (int x0 = tid * 4; x0 < X; x0 += TPB * 4) {
        float4 acc = make_float4(0.f, 0.f, 0.f, 0.f);
        for (int yy = 0; yy < Y; ++yy) {
            const float  yv = ys[yy];
            const float4 w  = *(const float4*)(W + (size_t)yy * X + x0);
            acc.x = fmaf(yv, w.x, acc.x);
            acc.y = fmaf(yv, w.y, acc.y);
            acc.z = fmaf(yv, w.z, acc.z);
            acc.w = fmaf(yv, w.w, acc.w);
        }
        const float4 bv = *(const float4*)(bias + (size_t)a * X + x0);
        acc.x += bv.x; acc.y += bv.y; acc.z += bv.z; acc.w += bv.w;
        *(float4*)(Wy + (size_t)b * X + x0) = acc;
    }
}

// ---------------------------------------------------------------------------
// Phase B: xWy[b,l] = dot(x[b,l,:], Wy[b,:]); mask; log_softmax over l.
// One block (8 waves) per batch; streams the batch's 4MB x slice exactly once.
// Each wave computes a 16-row l-tile via V_WMMA_F32_16X16X4_F32:
//   A (16x4 f32, ISA lane layout): lanes 0-15 hold K=0,1; lanes 16-31 K=2,3
//     -> per-lane contiguous 8B global load of x.
//   B (4x16): B[k,n] = Wy[k0+k] broadcast over n; same half-wave K split
//     -> per-lane 8B LDS load. D[m,n] is constant over n; lanes n==0 of each
//     half extract rows m=0..7 / 8..15 from the 8 accumulator VGPRs.
// ---------------------------------------------------------------------------
__global__ void __launch_bounds__(TPB) attn_kernel(
    const float* __restrict__ x, const unsigned char* __restrict__ x_mask,
    const float* __restrict__ Wy, float* __restrict__ out, int L, int X)
{
    const int b    = blockIdx.x;
    const int tid  = threadIdx.x;
    const int lane = tid & 31;
    const int wave = tid >> 5;
    const int nwaves = TPB / 32;

    __shared__ float wy_s[1024];   // X <= 1024
    __shared__ float row_s[1024];  // L <= 1024
    __shared__ float red[TPB];

    for (int i = tid; i < X; i += TPB) wy_s[i] = Wy[(size_t)b * X + i];
    __syncthreads();

    const int m  = lane & 15;   // matrix row within tile
    const int hi = lane >> 4;   // half-wave selects K pair {0,1} vs {2,3}
    const int ntiles = L >> 4;

    for (int t = wave; t < ntiles; t += nwaves) {
        const int l0 = t << 4;
        const float* __restrict__ xrow =
            x + ((size_t)b * L + (size_t)(l0 + m)) * X + 2 * hi;

        v8f c = {0.f, 0.f, 0.f, 0.f, 0.f, 0.f, 0.f, 0.f};
#pragma unroll 8
        for (int k0 = 0; k0 < X; k0 += 4) {
            const v2f av = *(const v2f*)(xrow + k0);          // global b64
            const v2f bv = *(const v2f*)(&wy_s[k0 + 2 * hi]); // ds b64
            c = __builtin_amdgcn_wmma_f32_16x16x4_f32(
                    /*neg_a=*/false, av, /*neg_b=*/false, bv,
                    /*c_mod=*/(short)0, c, /*reuse_a=*/false, /*reuse_b=*/false);
        }
        if (m == 0) {  // column N=0 of each half-wave holds valid results
            const int base = l0 + hi * 8;
#pragma unroll
            for (int r = 0; r < 8; ++r) row_s[base + r] = c[r];
        }
    }
    __syncthreads();

    // mask + row max
    float lmax = -INFINITY;
    for (int i = tid; i < L; i += TPB) {
        float v = row_s[i];
        if (x_mask[(size_t)b * L + i]) v = -INFINITY;
        row_s[i] = v;
        lmax = fmaxf(lmax, v);
    }
    red[tid] = lmax; __syncthreads();
    for (int s = TPB / 2; s > 0; s >>= 1) {
        if (tid < s) red[tid] = fmaxf(red[tid], red[tid + s]);
        __syncthreads();
    }
    const float gmax = red[0];
    __syncthreads();

    // sum(exp(v - gmax))
    float lsum = 0.f;
    for (int i = tid; i < L; i += TPB) lsum += __expf(row_s[i] - gmax);
    red[tid] = lsum; __syncthreads();
    for (int s = TPB / 2; s > 0; s >>= 1) {
        if (tid < s) red[tid] += red[tid + s];
        __syncthreads();
    }
    const float lse = gmax + logf(red[0]);

    for (int i = tid; i < L; i += TPB)
        out[(size_t)b * L + i] = row_s[i] - lse;
}

extern "C" void kernel_launch(void* const* d_in, const int* in_sizes, int n_in,
                              void* d_out, int out_size, void* d_ws, size_t ws_size,
                              hipStream_t stream) {
    const float*         x       = (const float*)d_in[0];         // [B,L,X] f32
    const float*         y       = (const float*)d_in[1];         // [B,Y]   f32
    const unsigned char* x_mask  = (const unsigned char*)d_in[2]; // [B,L]   bool
    const int*           actions = (const int*)d_in[3];           // [B]     i32
    const float*         weight  = (const float*)d_in[4];         // [A,Y,X] f32
    const float*         bias    = (const float*)d_in[5];         // [A,X]   f32
    float*               out     = (float*)d_out;                 // [B,L]   f32

    const int B = in_sizes[3];
    const int L = in_sizes[2] / B;
    const int Y = in_sizes[1] / B;
    const int X = in_sizes[0] / (B * L);

    float* Wy = (float*)d_ws;  // [B,X] f32 scratch (512 KB for nominal shapes)

    wy_kernel<<<B, TPB, (size_t)Y * sizeof(float), stream>>>(
        y, actions, weight, bias, Wy, L, X, Y);
    attn_kernel<<<B, TPB, 0, stream>>>(x, x_mask, Wy, out, L, X);
}